// Shift2D_23450521436916
// MI455X (gfx1250) — compile-verified
//
#include <hip/hip_runtime.h>

// Shift2D (N=16, C=256, H=W=128, fp32) — bandwidth-bound: 512 MB mandatory traffic
// @ 23.3 TB/s ≈ 22 us; ~0.7 GFLOP -> 1.3 FLOP/B, far below any compute roofline,
// so WMMA is the wrong tool. The MI455X-specific win is the async data path:
// stage rows into LDS with global_load_async_to_lds_b128 (512 B per wave32 op,
// no VGPR staging, ASYNCcnt + s_wait_asynccnt), serve the 4x tap reuse from LDS,
// and stream write-once output with non-temporal 128-bit stores.

typedef __attribute__((ext_vector_type(4))) float v4f;

#define H_ 128
#define W_ 128
#define TR 32          // output rows per block; H_/TR = 4 tiles per plane
#define NTHREADS 256   // 8 waves (wave32)

__global__ __launch_bounds__(NTHREADS) void shift2d_kernel(
    const float* __restrict__ in, const float* __restrict__ wt,
    float* __restrict__ out, int C)
{
    __shared__ __align__(16) float smem[(TR + 1) * W_];

    const int tile  = blockIdx.x & 3;   // H_/TR == 4
    const int plane = blockIdx.x >> 2;  // n*C + c
    const int c     = plane & (C - 1);  // C is a power of two (256)
    const int r0    = tile * TR;
    const int tid   = threadIdx.x;

    // Per-channel shift parameters: block-uniform -> SGPRs (s_load + s_floor_f32).
    const float sy = wt[2 * c + 0];
    const float sx = wt[2 * c + 1];
    const int   dy = (int)floorf(sy);
    const int   dx = (int)floorf(sx);
    const float fy = sy - (float)dy;
    const float fx = sx - (float)dx;
    const float wx0 = 1.0f - fx, wx1 = fx;
    const float wy0 = 1.0f - fy, wy1 = fy;

    const float* planeSrc = in + (size_t)plane * (size_t)(H_ * W_);

    // ---- Stage rows [r0+dy, r0+dy+TR] into LDS via async global->LDS copies ----
    // 33 rows * 32 segments of 16 B = 1056 lane-ops across the block.
    // OOB rows: clamp the source address (finite garbage), masked to zero in compute.
    //
    // The LDS destination address is derived from &smem[...] itself:
    //  * low 32 bits of the generic address of an LDS object == LDS byte address
    //  * ptrtoint capture makes smem "escape", so the "memory" clobber forces the
    //    compiler to treat the asm as writing smem (otherwise the post-barrier
    //    reads fold to poison and the whole stencil is dead-code-eliminated).
    for (int seg = tid; seg < (TR + 1) * 32; seg += NTHREADS) {
        int rr = seg >> 5;             // staged row 0..TR
        int cb = (seg & 31) << 4;      // byte offset within the 512 B row
        int gy = r0 + dy + rr;
        gy = gy < 0 ? 0 : (gy > H_ - 1 ? H_ - 1 : gy);
        const char* src = (const char*)(planeSrc + (size_t)gy * W_) + cb;
        unsigned ldsoff = (unsigned)(unsigned long long)&smem[rr * W_ + (cb >> 2)];
        asm volatile("global_load_async_to_lds_b128 %0, %1, off"
                     :: "v"(ldsoff), "v"(src)
                     : "memory");
    }
    // Each wave drains its own ASYNCcnt, then the workgroup barrier publishes LDS.
    asm volatile("s_wait_asynccnt 0x0" ::: "memory");
    __syncthreads();

    // ---- Compute: each thread emits 4 consecutive outputs per iteration ----
    const int jbase  = (tid & 31) * 4;  // output column base (16 B aligned)
    const int rowSel = tid >> 5;        // 0..7
    float* planeDst = out + (size_t)plane * (size_t)(H_ * W_);

    #pragma unroll
    for (int it = 0; it < TR / 8; ++it) {
        const int il = rowSel + it * 8;       // local output row (0..TR-1)
        const int i  = r0 + il;               // global output row
        const int y0 = i + dy;
        const float my0 = ((unsigned)y0       < (unsigned)H_) ? 1.0f : 0.0f;
        const float my1 = ((unsigned)(y0 + 1) < (unsigned)H_) ? 1.0f : 0.0f;
        const float wy0m = wy0 * my0;
        const float wy1m = wy1 * my1;
        const int baseA = il * W_;            // staged row y0
        const int baseB = (il + 1) * W_;      // staged row y0 + 1

        // 5 shared column taps produce 4 outputs (tap k+1 reused as right tap of k).
        float vb[5];
        #pragma unroll
        for (int k = 0; k < 5; ++k) {
            int x = jbase + dx + k;
            float mx = ((unsigned)x < (unsigned)W_) ? 1.0f : 0.0f;
            int xc = x < 0 ? 0 : (x > W_ - 1 ? W_ - 1 : x);
            vb[k] = mx * (wy0m * smem[baseA + xc] + wy1m * smem[baseB + xc]);
        }

        v4f o;
        #pragma unroll
        for (int k = 0; k < 4; ++k)
            o[k] = wx0 * vb[k] + wx1 * vb[k + 1];

        __builtin_nontemporal_store(o, (v4f*)(planeDst + (size_t)i * W_ + jbase));
    }
}

extern "C" void kernel_launch(void* const* d_in, const int* in_sizes, int n_in,
                              void* d_out, int out_size, void* d_ws, size_t ws_size,
                              hipStream_t stream) {
    const float* in = (const float*)d_in[0];   // (16, 256, 128, 128) fp32
    const float* wt = (const float*)d_in[1];   // (256, 2) fp32
    float* out = (float*)d_out;

    const int C = 256;
    const int planes = in_sizes[0] / (H_ * W_);     // 16*256 = 4096
    dim3 grid(planes * (H_ / TR));                   // 16384 blocks
    shift2d_kernel<<<grid, NTHREADS, 0, stream>>>(in, wt, out, C);
}